// Router_15436112461974
// MI455X (gfx1250) — compile-verified
//
#include <hip/hip_runtime.h>
#include <hip/hip_bf16.h>

#define NUM_EXPERTS 64
#define D_MODEL     2048
#define NUM_GROUPS  16
#define GROUP_SIZE  1024
#define TOKENS      (NUM_GROUPS * GROUP_SIZE)   // 16384
#define BLK_TOK     32                          // tokens per workgroup (2 waves x 16)

// d_ws layout (bytes):
//   [0]            float z2sum
//   [4..259]       float usage[64]
//   [4096 ..)      float Wt[64][2048]     (transposed gate weights, 512 KB)
//   [528384 ..)    float probs[64][16384] (4 MB)
#define WS_WT_OFF    4096
#define WS_PROBS_OFF (WS_WT_OFF + NUM_EXPERTS * D_MODEL * 4)

typedef __attribute__((ext_vector_type(2))) float v2f;
typedef __attribute__((ext_vector_type(8))) float v8f;

// ---------------------------------------------------------------------------
// Kernel 0: transpose W (D_MODEL, E) -> Wt (E, D_MODEL) so the GEMM's B
// fragments are contiguous along K (one b64 load per lane per k-step).
// ---------------------------------------------------------------------------
__global__ __launch_bounds__(256)
void transpose_w(const float* __restrict__ W, float* __restrict__ Wt)
{
    int idx = blockIdx.x * 256 + threadIdx.x;   // 131072 threads
    int e = idx >> 11;                          // / D_MODEL
    int k = idx & (D_MODEL - 1);
    Wt[idx] = W[(size_t)k * NUM_EXPERTS + e];
}

// ---------------------------------------------------------------------------
// Kernel 1: logits GEMM (fp32 WMMA 16x16x4) + fused softmax + loss partials
// Each wave: 16 tokens x 64 experts (4 WMMA tiles), K-loop step 4, unroll 8.
// ---------------------------------------------------------------------------
__global__ __launch_bounds__(64)
void router_gemm_softmax(const float* __restrict__ x,
                         const float* __restrict__ Wt,
                         float* __restrict__ probs,
                         float* __restrict__ usage,
                         float* __restrict__ z2sum)
{
    __shared__ float lds[NUM_EXPERTS * (BLK_TOK + 1)];   // stride 33 (odd) -> no bank conflicts

    const int tid  = threadIdx.x;
    const int wave = tid >> 5;          // 0..1
    const int lane = tid & 31;
    const int half = lane >> 4;         // 0: lanes 0-15, 1: lanes 16-31
    const int l    = lane & 15;

    const int blockTok = blockIdx.x * BLK_TOK;
    const int tBase    = blockTok + wave * 16;

    v8f acc[4];
    for (int j = 0; j < 4; ++j)
        for (int i = 0; i < 8; ++i) acc[j][i] = 0.0f;

    // A-fragment rows: M = l (same 16 rows for both lane halves)
    const float* xrow = x + (size_t)(tBase + l) * D_MODEL;
    // B-fragment columns: N = l, one pointer per 16-wide expert tile
    const float* wt0 = Wt + (size_t)(16 * 0 + l) * D_MODEL;
    const float* wt1 = Wt + (size_t)(16 * 1 + l) * D_MODEL;
    const float* wt2 = Wt + (size_t)(16 * 2 + l) * D_MODEL;
    const float* wt3 = Wt + (size_t)(16 * 3 + l) * D_MODEL;

    #pragma unroll 8
    for (int k0 = 0; k0 < D_MODEL; k0 += 4) {
        const int kk = k0 + 2 * half;            // layout: v0 = K0|K2, v1 = K1|K3
        v2f a, b0, b1, b2, b3;
        a.x  = xrow[kk]; a.y  = xrow[kk + 1];
        b0.x = wt0[kk];  b0.y = wt0[kk + 1];
        b1.x = wt1[kk];  b1.y = wt1[kk + 1];
        b2.x = wt2[kk];  b2.y = wt2[kk + 1];
        b3.x = wt3[kk];  b3.y = wt3[kk + 1];
        acc[0] = __builtin_amdgcn_wmma_f32_16x16x4_f32(false, a, false, b0, (short)0, acc[0], false, false);
        acc[1] = __builtin_amdgcn_wmma_f32_16x16x4_f32(false, a, false, b1, (short)0, acc[1], false, false);
        acc[2] = __builtin_amdgcn_wmma_f32_16x16x4_f32(false, a, false, b2, (short)0, acc[2], false, false);
        acc[3] = __builtin_amdgcn_wmma_f32_16x16x4_f32(false, a, false, b3, (short)0, acc[3], false, false);
    }

    // Fused softmax over 64 experts per token.
    // C/D layout: VGPR r -> token M=r (lanes 0-15) / M=r+8 (lanes 16-31), expert = 16*j + (lane&15).
    // xor masks < 16 keep reductions inside each 16-lane half.
    #pragma unroll
    for (int r = 0; r < 8; ++r) {
        float v0 = acc[0][r], v1 = acc[1][r], v2 = acc[2][r], v3 = acc[3][r];
        float m = fmaxf(fmaxf(v0, v1), fmaxf(v2, v3));
        m = fmaxf(m, __shfl_xor(m, 1));
        m = fmaxf(m, __shfl_xor(m, 2));
        m = fmaxf(m, __shfl_xor(m, 4));
        m = fmaxf(m, __shfl_xor(m, 8));
        float p0 = __expf(v0 - m), p1 = __expf(v1 - m);
        float p2 = __expf(v2 - m), p3 = __expf(v3 - m);
        float s = p0 + p1 + p2 + p3;
        s += __shfl_xor(s, 1);
        s += __shfl_xor(s, 2);
        s += __shfl_xor(s, 4);
        s += __shfl_xor(s, 8);
        float z = m + __logf(s);                 // log-sum-exp (router_z)
        if (l == 0) atomicAdd(z2sum, z * z);     // lanes 0 & 16: tokens r and r+8
        float inv = 1.0f / s;
        int tokLocal = wave * 16 + r + 8 * half; // 0..31
        lds[(16 * 0 + l) * (BLK_TOK + 1) + tokLocal] = p0 * inv;
        lds[(16 * 1 + l) * (BLK_TOK + 1) + tokLocal] = p1 * inv;
        lds[(16 * 2 + l) * (BLK_TOK + 1) + tokLocal] = p2 * inv;
        lds[(16 * 3 + l) * (BLK_TOK + 1) + tokLocal] = p3 * inv;
    }
    __syncthreads();

    // Coalesced (E, T) store of probabilities.
    for (int idx = tid; idx < NUM_EXPERTS * BLK_TOK; idx += 64) {
        int e = idx >> 5;           // /32
        int t = idx & (BLK_TOK - 1);
        probs[(size_t)e * TOKENS + blockTok + t] = lds[e * (BLK_TOK + 1) + t];
    }
    // Per-expert usage partial sums (thread e sums this block's 32 tokens).
    {
        float s = 0.0f;
        for (int t = 0; t < BLK_TOK; ++t) s += lds[tid * (BLK_TOK + 1) + t];
        atomicAdd(usage + tid, s);
    }
}

// ---------------------------------------------------------------------------
// Kernel 2: per-expert exact top-k (radix select on float bits, probs >= 0)
// + scatter into expert/weight masks. One workgroup per expert, the expert's
// 16384 probs cached in 64 KB of LDS (dynamic shared).
// Ties at the threshold are taken in ascending token order (top_k semantics).
// ---------------------------------------------------------------------------
__global__ __launch_bounds__(256)
void topk_scatter(const float* __restrict__ probs,
                  const int* __restrict__ capP,
                  float* __restrict__ maskOut,
                  float* __restrict__ weightOut)
{
    extern __shared__ float sprobs[];      // TOKENS floats = 64 KB
    __shared__ int hist[256];
    __shared__ int scanBuf[256];
    __shared__ int sRem, sBin;

    const int tid = threadIdx.x;
    const int e   = blockIdx.x;
    const float* pe = probs + (size_t)e * TOKENS;

    for (int t = tid; t < TOKENS; t += 256) sprobs[t] = pe[t];
    if (tid == 0) {
        int k = *capP;
        if (k > TOKENS) k = TOKENS;
        if (k < 1) k = 1;
        sRem = k;
    }
    __syncthreads();

    unsigned prefix_val = 0u, prefix_mask = 0u;
    for (int pass = 0; pass < 4; ++pass) {
        const int shift = 24 - 8 * pass;
        hist[tid] = 0;
        __syncthreads();
        for (int t = tid; t < TOKENS; t += 256) {
            unsigned key = __float_as_uint(sprobs[t]);
            if ((key & prefix_mask) == prefix_val)
                atomicAdd(&hist[(key >> shift) & 255u], 1);
        }
        __syncthreads();
        if (tid == 0) {
            int rem = sRem;
            int cum = 0, b = 255;
            for (; b >= 0; --b) {
                cum += hist[b];
                if (cum >= rem) break;
            }
            if (b < 0) b = 0;                    // safety; cannot happen
            sBin = b;
            sRem = rem - (cum - hist[b]);        // ties still needed inside bin b
        }
        __syncthreads();
        prefix_val  |= ((unsigned)sBin) << shift;
        prefix_mask |= 0xFFu << shift;
        __syncthreads();
    }

    const unsigned thresh = prefix_val;          // exact key of threshold element
    const int tiesNeeded  = sRem;                // # of ==thresh elements to keep

    // Blocked tie ranking: thread owns tokens [tid*64, tid*64+64)
    const int base = tid * 64;
    int localTies = 0;
    for (int i = 0; i < 64; ++i)
        localTies += (__float_as_uint(sprobs[base + i]) == thresh) ? 1 : 0;
    scanBuf[tid] = localTies;
    __syncthreads();
    for (int off = 1; off < 256; off <<= 1) {    // Hillis-Steele inclusive scan
        int add = (tid >= off) ? scanBuf[tid - off] : 0;
        __syncthreads();
        scanBuf[tid] += add;
        __syncthreads();
    }
    int rank = scanBuf[tid] - localTies;         // exclusive prefix of ties

    float* me = maskOut   + (size_t)e * TOKENS;
    float* we = weightOut + (size_t)e * TOKENS;
    for (int i = 0; i < 64; ++i) {
        int t = base + i;
        float p = sprobs[t];
        unsigned key = __float_as_uint(p);
        bool sel = false;
        if (key > thresh) sel = true;
        else if (key == thresh) { sel = (rank < tiesNeeded); ++rank; }
        if (sel) { me[t] = 1.0f; we[t] = p; }
    }
}

// ---------------------------------------------------------------------------
// Kernel 3: finalize scalar loss.
// ---------------------------------------------------------------------------
__global__ __launch_bounds__(64)
void finalize_loss(const float* __restrict__ z2sum,
                   const float* __restrict__ usage,
                   float* __restrict__ lossOut)
{
    __shared__ float red[64];
    const int tid = threadIdx.x;
    float u = usage[tid] * (1.0f / (float)TOKENS);
    red[tid] = u * u;
    __syncthreads();
    for (int off = 32; off > 0; off >>= 1) {
        if (tid < off) red[tid] += red[tid + off];
        __syncthreads();
    }
    if (tid == 0) {
        float bal = ((float)NUM_EXPERTS * red[0] - 1.0f) * 0.01f;
        float zl  = (z2sum[0] * (1.0f / (float)TOKENS)) * 0.001f;
        lossOut[0] = bal + zl;
    }
}

// ---------------------------------------------------------------------------
extern "C" void kernel_launch(void* const* d_in, const int* in_sizes, int n_in,
                              void* d_out, int out_size, void* d_ws, size_t ws_size,
                              hipStream_t stream)
{
    const float* x   = (const float*)d_in[0];
    const float* W   = (const float*)d_in[1];
    const int*   cap = (const int*)d_in[2];

    float* out     = (float*)d_out;
    float* masks   = out;                                   // (64,16,1024) as 0/1 floats
    float* weights = out + (size_t)NUM_EXPERTS * TOKENS;    // (64,16,1024)
    float* loss    = out + 2 * (size_t)NUM_EXPERTS * TOKENS;

    float* z2    = (float*)d_ws;                            // [0]
    float* usage = (float*)d_ws + 1;                        // [1..64]
    float* Wt    = (float*)((char*)d_ws + WS_WT_OFF);       // (64, 2048)
    float* probs = (float*)((char*)d_ws + WS_PROBS_OFF);    // (64, 16384)

    hipMemsetAsync(d_out, 0, (size_t)out_size * sizeof(float), stream);
    hipMemsetAsync(d_ws, 0, WS_WT_OFF, stream);

    transpose_w<<<(NUM_EXPERTS * D_MODEL) / 256, 256, 0, stream>>>(W, Wt);
    router_gemm_softmax<<<TOKENS / BLK_TOK, 64, 0, stream>>>(x, Wt, probs, usage, z2);
    topk_scatter<<<NUM_EXPERTS, 256, TOKENS * sizeof(float), stream>>>(probs, cap, masks, weights);
    finalize_loss<<<1, 64, 0, stream>>>(z2, usage, loss);
}